// CTCBeamSearch_8735963480192
// MI455X (gfx1250) — compile-verified
//
#include <hip/hip_runtime.h>
#include <math.h>

#define VOCAB     50257
#define BEAM      32
#define NBINS     2048
#define CAND_CAP  256
#define P1_THREADS 256

typedef float v2f __attribute__((ext_vector_type(2)));
typedef float v8f __attribute__((ext_vector_type(8)));

// Monotonic float -> unsigned key (order-preserving for all finite floats).
__device__ __forceinline__ unsigned fkey(float x) {
    unsigned u = __float_as_uint(x);
    return u ^ ((u >> 31) ? 0xFFFFFFFFu : 0x80000000u);
}

// Total order used by top_k: larger value first; ties -> smaller index first.
__device__ __forceinline__ bool better(float av, int ai, float bv, int bi) {
    return (av > bv) || (av == bv && ai < bi);
}

// ---------------------------------------------------------------------------
// Phase 1: one workgroup per frame. Computes logsumexp and the top-32
// (value,index) of the frame via histogram-thresholding + LDS bitonic sort.
// Memory-bound: two streaming passes over the 201KB row, bulk as b128 loads.
// ---------------------------------------------------------------------------
__global__ void ctc_phase1_topk_lse(const float* __restrict__ enc,
                                    float* __restrict__ topv,
                                    int* __restrict__ topi) {
    const int t   = blockIdx.x;
    const int tid = threadIdx.x;
    const float* row = enc + (size_t)t * VOCAB;

    // Row base is only 4B-aligned (stride 50257 floats); split head/vec/tail.
    const int head = (int)((16u - ((unsigned)(uintptr_t)row & 15u)) & 15u) >> 2; // 0..3
    const int nvec = (VOCAB - head) >> 2;
    const float4* __restrict__ row4 = (const float4*)(row + head);
    const int tailStart = head + (nvec << 2);

    __shared__ unsigned hist[NBINS];
    __shared__ float    red[P1_THREADS];
    __shared__ float    candV[CAND_CAP];
    __shared__ int      candI[CAND_CAP];
    __shared__ unsigned s_threshKey;
    __shared__ int      s_cnt;
    __shared__ float    s_M, s_lse;

    for (int i = tid; i < NBINS; i += P1_THREADS) hist[i] = 0u;
    if (tid == 0) s_cnt = 0;
    __syncthreads();

    // ---- Pass A: running max + 2048-bin histogram of ordered keys ----
    float m = -3.0e38f;
    for (int i = tid; i < nvec; i += P1_THREADS) {
        float4 x = row4[i];
        m = fmaxf(m, fmaxf(fmaxf(x.x, x.y), fmaxf(x.z, x.w)));
        atomicAdd(&hist[fkey(x.x) >> 21], 1u);
        atomicAdd(&hist[fkey(x.y) >> 21], 1u);
        atomicAdd(&hist[fkey(x.z) >> 21], 1u);
        atomicAdd(&hist[fkey(x.w) >> 21], 1u);
    }
    if (tid < head) {                           // head scalars (0..3)
        float x = row[tid];
        m = fmaxf(m, x);
        atomicAdd(&hist[fkey(x) >> 21], 1u);
    }
    {
        int i = tailStart + tid;                // tail scalars (0..3)
        if (i < VOCAB) {
            float x = row[i];
            m = fmaxf(m, x);
            atomicAdd(&hist[fkey(x) >> 21], 1u);
        }
    }
    red[tid] = m;
    __syncthreads();
    for (int st = P1_THREADS >> 1; st > 0; st >>= 1) {
        if (tid < st) red[tid] = fmaxf(red[tid], red[tid + st]);
        __syncthreads();
    }
    if (tid == 0) {
        s_M = red[0];
        unsigned cum = 0; int b = NBINS;
        while (b > 0 && cum < BEAM) { --b; cum += hist[b]; }
        s_threshKey = ((unsigned)b) << 21;
    }
    __syncthreads();
    const float    M  = s_M;
    const unsigned tk = s_threshKey;

    // ---- Pass B: sum(exp(x-M)) + collect candidates >= threshold ----
    float sum = 0.0f;
    for (int i = tid; i < nvec; i += P1_THREADS) {
        float4 x = row4[i];
        sum += expf(x.x - M) + expf(x.y - M) + expf(x.z - M) + expf(x.w - M);
        const int base = head + (i << 2);
        if (fkey(x.x) >= tk) { int p = atomicAdd(&s_cnt, 1); if (p < CAND_CAP) { candV[p] = x.x; candI[p] = base + 0; } }
        if (fkey(x.y) >= tk) { int p = atomicAdd(&s_cnt, 1); if (p < CAND_CAP) { candV[p] = x.y; candI[p] = base + 1; } }
        if (fkey(x.z) >= tk) { int p = atomicAdd(&s_cnt, 1); if (p < CAND_CAP) { candV[p] = x.z; candI[p] = base + 2; } }
        if (fkey(x.w) >= tk) { int p = atomicAdd(&s_cnt, 1); if (p < CAND_CAP) { candV[p] = x.w; candI[p] = base + 3; } }
    }
    if (tid < head) {
        float x = row[tid];
        sum += expf(x - M);
        if (fkey(x) >= tk) { int p = atomicAdd(&s_cnt, 1); if (p < CAND_CAP) { candV[p] = x; candI[p] = tid; } }
    }
    {
        int i = tailStart + tid;
        if (i < VOCAB) {
            float x = row[i];
            sum += expf(x - M);
            if (fkey(x) >= tk) { int p = atomicAdd(&s_cnt, 1); if (p < CAND_CAP) { candV[p] = x; candI[p] = i; } }
        }
    }
    red[tid] = sum;
    __syncthreads();
    int cnt = s_cnt; cnt = (cnt < CAND_CAP) ? cnt : CAND_CAP;
    if (tid >= cnt) { candV[tid] = -3.0e38f; candI[tid] = 0x7FFFFFFF; }
    for (int st = P1_THREADS >> 1; st > 0; st >>= 1) {
        if (tid < st) red[tid] += red[tid + st];
        __syncthreads();
    }
    if (tid == 0) s_lse = M + logf(red[0]);
    __syncthreads();

    // ---- Bitonic sort of 256 slots: slot 0 = best (value desc, index asc) ----
    for (unsigned k = 2; k <= CAND_CAP; k <<= 1) {
        for (unsigned j = k >> 1; j > 0; j >>= 1) {
            unsigned ixj = (unsigned)tid ^ j;
            if (ixj > (unsigned)tid) {
                float av = candV[tid]; int ai = candI[tid];
                float bv = candV[ixj]; int bi = candI[ixj];
                bool asc = ((tid & k) == 0);
                bool sw  = asc ? better(bv, bi, av, ai) : better(av, ai, bv, bi);
                if (sw) {
                    candV[tid] = bv; candI[tid] = bi;
                    candV[ixj] = av; candI[ixj] = ai;
                }
            }
            __syncthreads();
        }
    }

    if (tid < BEAM) {
        topv[(size_t)t * BEAM + tid] = candV[tid] - s_lse;  // log-softmax value
        topi[(size_t)t * BEAM + tid] = candI[tid];
    }
}

// ---------------------------------------------------------------------------
// Phase 2: single wave. Per timestep, build cand[b][j] = s[b] + L[j] as a
// rank-2 WMMA update (4x v_wmma_f32_16x16x4_f32 tiles -> LDS), then extract
// the top-32 with a 32-way merge (rows are sorted desc), using butterfly
// argmax with JAX-compatible flat-index tie-breaking.
// ---------------------------------------------------------------------------
__global__ void ctc_phase2_beam(const float* __restrict__ topv,
                                const int* __restrict__ topi,
                                float* __restrict__ out, int T) {
    const int lane = threadIdx.x;            // 0..31, wave32
    __shared__ float cand[32][33];           // padded rows

    float s = (lane == 0) ? 0.0f : -1.0e30f; // _init_hypos: one live hypothesis
    float* outScores = out;
    float* outTok    = out + BEAM;
    float* outHyp    = out + BEAM + (size_t)T * BEAM;

    for (int t = 0; t < T; ++t) {
        float L = topv[(size_t)t * BEAM + lane];   // sorted desc
        int   I = topi[(size_t)t * BEAM + lane];

        const int lo = lane & 15;
        const int hi = lane >> 4;

        // cand = s * 1^T + 1 * L^T via D = A(16x4) x B(4x16) + 0
        #pragma unroll
        for (int tm = 0; tm < 2; ++tm) {
            float sA = __shfl(s, tm * 16 + lo);
            #pragma unroll
            for (int tn = 0; tn < 2; ++tn) {
                float lB = __shfl(L, tn * 16 + lo);
                v2f a, b;
                a.x = (hi == 0) ? sA   : 0.0f;  // A: K=0 | K=2
                a.y = (hi == 0) ? 1.0f : 0.0f;  //    K=1 | K=3
                b.x = (hi == 0) ? 1.0f : 0.0f;  // B: K=0 | K=2
                b.y = (hi == 0) ? lB   : 0.0f;  //    K=1 | K=3
                v8f c = {};
                v8f d = __builtin_amdgcn_wmma_f32_16x16x4_f32(
                    false, a, false, b, (short)0, c, false, false);
                int colBase = tn * 16 + lo;
                int rowBase = tm * 16 + hi * 8;
                #pragma unroll
                for (int r = 0; r < 8; ++r)
                    cand[rowBase + r][colBase] = d[r];
            }
        }
        __syncthreads();

        // 32-way merge: lane b owns row b with pointer j.
        int   j      = 0;
        float curV   = cand[lane][0];
        int   curTok = __shfl(I, 0);
        float myS = 0.0f; int myT = 0, myH = 0;

        #pragma unroll 1
        for (int k = 0; k < BEAM; ++k) {
            float v    = curV;
            int   flat = lane * VOCAB + curTok;      // < 2^24, exact
            #pragma unroll
            for (int off = 16; off > 0; off >>= 1) {
                float ov = __shfl_xor(v, off);
                int   of = __shfl_xor(flat, off);
                if (ov > v || (ov == v && of < flat)) { v = ov; flat = of; }
            }
            int winB = flat / VOCAB;
            int winT = flat - winB * VOCAB;
            if (lane == k) { myS = v; myT = winT; myH = winB; }

            bool adv = (lane == winB);
            if (adv) ++j;
            int   jj   = (j < BEAM) ? j : (BEAM - 1);
            float nv   = cand[lane][jj];
            int   ntok = __shfl(I, jj);              // uniform shfl, varying index
            if (adv) {
                curV   = (j < BEAM) ? nv : -3.0e38f;
                curTok = ntok;
            }
        }
        s = myS;                                     // new beam scores, sorted desc
        outTok[(size_t)t * BEAM + lane] = (float)myT;
        outHyp[(size_t)t * BEAM + lane] = (float)myH;
        __syncthreads();
    }
    outScores[lane] = s;
}

extern "C" void kernel_launch(void* const* d_in, const int* in_sizes, int n_in,
                              void* d_out, int out_size, void* d_ws, size_t ws_size,
                              hipStream_t stream) {
    (void)n_in; (void)out_size; (void)ws_size;
    const float* enc = (const float*)d_in[0];
    const int T = in_sizes[0] / VOCAB;   // 2048

    float* topv = (float*)d_ws;                                  // T*32 floats
    int*   topi = (int*)((char*)d_ws + (size_t)T * BEAM * sizeof(float));

    ctc_phase1_topk_lse<<<T, P1_THREADS, 0, stream>>>(enc, topv, topi);
    ctc_phase2_beam<<<1, BEAM, 0, stream>>>(topv, topi, (float*)d_out, T);
}